// so3_to_mat_31421980737562
// MI455X (gfx1250) — compile-verified
//
#include <hip/hip_runtime.h>
#include <stdint.h>

// ---------------------------------------------------------------------------
// SE(3) exp map: x[B,6] = (axis-angle w, translation t) -> out[B,4,4]
// Memory-bound: 96MB in + 256MB out => ~15us floor @ 23.3 TB/s.
// Strategy: TDM (tensor_load_to_lds) streams each block's 6KB input tile into
// LDS (linear DRAM reads, vector-mem pipe stays free), lanes read 6 floats
// from LDS, compute R = (1-B*th2)*I + A*K + B*w*w^T, and emit 4 coalesced
// non-temporal b128 stores per element.
// ---------------------------------------------------------------------------

typedef float        f32x4    __attribute__((ext_vector_type(4)));
typedef unsigned int uint32x4 __attribute__((ext_vector_type(4)));
typedef int          int32x8  __attribute__((ext_vector_type(8)));
typedef int          int32x4  __attribute__((ext_vector_type(4)));

#define TILE 256

#if __has_builtin(__builtin_amdgcn_tensor_load_to_lds) && __has_builtin(__builtin_amdgcn_s_wait_tensorcnt)
#define USE_TDM 1
#else
#define USE_TDM 0
#endif

#if USE_TDM
// LDS byte offset of a __shared__ object (generic -> AS(3) -> integer).
__device__ __forceinline__ unsigned lds_addr_of(const void* p) {
    return (unsigned)(uintptr_t)(const __attribute__((address_space(3))) void*)p;
}
#endif

__global__ __launch_bounds__(TILE) void se3_exp_kernel(const float* __restrict__ x,
                                                       float* __restrict__ out,
                                                       int n) {
    __shared__ float lds_in[TILE * 6];

    const int block0 = blockIdx.x * TILE;
    const int elems  = min(TILE, n - block0);   // last block may be partial
    const int t      = threadIdx.x;

#if USE_TDM
    if (threadIdx.x < 32) {  // exactly wave 0 issues the DMA (uniform branch)
        const unsigned nflt  = (unsigned)(elems * 6);            // <= 1536, fits tile_dim0[15:0]
        const uint64_t gaddr = (uint64_t)(const void*)(x + (size_t)block0 * 6);
        const unsigned laddr = lds_addr_of(lds_in);

        // ---- D# group 0 (128b): count=1 | lds_addr | global_addr | type=2
        uint32x4 g0;
        g0.x = 1u;                                               // count=1, user mode, no gather
        g0.y = laddr;                                            // lds_addr (bytes)
        g0.z = (unsigned)(gaddr & 0xffffffffu);                  // global_addr[31:0]
        g0.w = (unsigned)((gaddr >> 32) & 0x01ffffffu) | (2u << 30); // addr[56:32] | type=2

        // ---- D# group 1 (256b): data_size=4B, 1-row tile of nflt floats
        int32x8 g1;
        g1[0] = (int)(2u << 16);                                 // wg_mask=0, data_size=2 (4B)
        g1[1] = (int)((nflt & 0xffffu) << 16);                   // tensor_dim0[15:0] @ bits63:48
        g1[2] = (int)(((nflt >> 16) & 0xffffu) | (1u << 16));    // tensor_dim0[31:16] | tensor_dim1=1
        g1[3] = (int)((nflt & 0xffffu) << 16);                   // tensor_dim1[31:16]=0 | tile_dim0=nflt
        g1[4] = 1;                                               // tile_dim1=1, tile_dim2=0
        g1[5] = (int)nflt;                                       // tensor_dim0_stride[31:0]
        g1[6] = (int)((nflt & 0xffffu) << 16);                   // stride0[47:32]=0 | stride1[15:0]
        g1[7] = (int)((nflt >> 16) & 0xffffu);                   // stride1[47:16]

        int32x4 g2 = {0, 0, 0, 0};                               // tensor_dim2.. unused
        int32x4 g3 = {0, 0, 0, 0};
        int32x8 g4 = {0, 0, 0, 0, 0, 0, 0, 0};                   // 6-arg toolchain extra group

        __builtin_amdgcn_tensor_load_to_lds(g0, g1, g2, g3, g4, /*cpol=*/0);
        __builtin_amdgcn_s_wait_tensorcnt(0);                    // DMA landed in LDS
    }
    __syncthreads();
#endif

    if (t < elems) {
        const int i = block0 + t;

        float wx, wy, wz, tx, ty, tz;
#if USE_TDM
        const float* p = &lds_in[6 * t];
        wx = p[0]; wy = p[1]; wz = p[2];
        tx = p[3]; ty = p[4]; tz = p[5];
#else
        const float* p = x + (size_t)i * 6;
        wx = p[0]; wy = p[1]; wz = p[2];
        tx = p[3]; ty = p[4]; tz = p[5];
#endif

        const float th2 = wx * wx + wy * wy + wz * wz;

        float A, Bc;
        if (th2 < 1e-8f) {                      // Taylor branch (rare)
            A  = 1.0f - th2 * (1.0f / 6.0f);
            Bc = 0.5f - th2 * (1.0f / 24.0f);
        } else {
            const float theta = sqrtf(th2);
            float s, c;
            sincosf(theta, &s, &c);
            A  = s / theta;
            Bc = (1.0f - c) / th2;
        }

        // R = I + A*K + B*K^2, with K^2 = w*w^T - th2*I:
        // R = (1 - B*th2)*I + A*K + B*w*w^T  (diagonal folded below)
        const float bxy = Bc * wx * wy;
        const float bxz = Bc * wx * wz;
        const float byz = Bc * wy * wz;
        const float awx = A * wx;
        const float awy = A * wy;
        const float awz = A * wz;

        f32x4 r0 = { 1.0f - Bc * (wy * wy + wz * wz), bxy - awz, bxz + awy, tx };
        f32x4 r1 = { bxy + awz, 1.0f - Bc * (wx * wx + wz * wz), byz - awx, ty };
        f32x4 r2 = { bxz - awy, byz + awx, 1.0f - Bc * (wx * wx + wy * wy), tz };
        f32x4 r3 = { 0.0f, 0.0f, 0.0f, 1.0f };

        // 64B/element, write-once stream > L2 capacity -> non-temporal b128 stores.
        f32x4* o = (f32x4*)(out + (size_t)i * 16);
        __builtin_nontemporal_store(r0, o + 0);
        __builtin_nontemporal_store(r1, o + 1);
        __builtin_nontemporal_store(r2, o + 2);
        __builtin_nontemporal_store(r3, o + 3);
    }
}

extern "C" void kernel_launch(void* const* d_in, const int* in_sizes, int n_in,
                              void* d_out, int out_size, void* d_ws, size_t ws_size,
                              hipStream_t stream) {
    (void)n_in; (void)out_size; (void)d_ws; (void)ws_size;
    const float* x   = (const float*)d_in[0];
    float*       out = (float*)d_out;
    const int n = in_sizes[0] / 6;            // 4,000,000
    if (n <= 0) return;
    const int blocks = (n + TILE - 1) / TILE; // 15,625
    se3_exp_kernel<<<blocks, TILE, 0, stream>>>(x, out, n);
}